// SECOND_BEAM_SEARCH_18923625906730
// MI455X (gfx1250) — compile-verified
//
#include <hip/hip_runtime.h>
#include <hip/hip_bf16.h>
#include <stdint.h>

typedef float v4f __attribute__((ext_vector_type(4)));

#define L_     16
#define BEAM_  3
#define H_     8
#define S_     2048
#define D_     128
#define VOCAB_ 32000
#define HIST_  128
#define TOPK_  3

static constexpr long KVN    = (long)L_ * BEAM_ * H_ * S_ * D_;   // 100663296
static constexpr long O_TOK  = KVN;                               // tok_col (3)
static constexpr long O_SAVE = O_TOK + BEAM_;                     // save_id_out (3*129)
static constexpr long O_RP   = O_SAVE + (long)BEAM_ * (HIST_ + 1);// rp (3*32000)
static constexpr long O_PROB = O_RP + (long)BEAM_ * VOCAB_;       // top_beam_prob (3)
static constexpr long O_MAXI = O_PROB + BEAM_;                    // max_logits_idx (1)

// ---------------------------------------------------------------------------
// Kernel A: log-softmax + per-beam top-3 + beam combine + small outputs
// ---------------------------------------------------------------------------
#define TPB_A 1024

__device__ __forceinline__ void ins3(float x, int idx, float v[3], int ii[3]) {
    if (x > v[0] || (x == v[0] && idx < ii[0])) {
        v[2] = v[1]; ii[2] = ii[1];
        v[1] = v[0]; ii[1] = ii[0];
        v[0] = x;    ii[0] = idx;
    } else if (x > v[1] || (x == v[1] && idx < ii[1])) {
        v[2] = v[1]; ii[2] = ii[1];
        v[1] = x;    ii[1] = idx;
    } else if (x > v[2] || (x == v[2] && idx < ii[2])) {
        v[2] = x;    ii[2] = idx;
    }
}

// merge two descending-sorted 3-lists; result (top-3) left in (av, ai)
__device__ __forceinline__ void merge3(float av[3], int ai[3],
                                       const float bv[3], const int bi2[3]) {
    float rv[3]; int ri[3];
    int pa = 0, pb = 0;
#pragma unroll
    for (int k = 0; k < 3; ++k) { // pa+pb == k <= 2, no bounds check needed
        bool ta = (av[pa] > bv[pb]) || (av[pa] == bv[pb] && ai[pa] <= bi2[pb]);
        if (ta) { rv[k] = av[pa]; ri[k] = ai[pa]; ++pa; }
        else    { rv[k] = bv[pb]; ri[k] = bi2[pb]; ++pb; }
    }
#pragma unroll
    for (int k = 0; k < 3; ++k) { av[k] = rv[k]; ai[k] = ri[k]; }
}

__global__ __launch_bounds__(TPB_A) void beam_topk_kernel(
    const float* __restrict__ logits, const int* __restrict__ save_id,
    const float* __restrict__ rp_in, const float* __restrict__ prev,
    float* __restrict__ out, int* __restrict__ ws)
{
    __shared__ float s_red[TPB_A];
    __shared__ float sv[TPB_A * 3];
    __shared__ int   si[TPB_A * 3];
    __shared__ float btv[BEAM_][TOPK_];
    __shared__ int   bti[BEAM_][TOPK_];
    __shared__ int   s_bi[BEAM_], s_tok[BEAM_];

    const int   tid = threadIdx.x;
    const float NEG = -3.4e38f;

    for (int b = 0; b < BEAM_; ++b) {
        const float* lb = logits + (long)b * VOCAB_;
        const float* rb = rp_in  + (long)b * VOCAB_;

        // pass 1: thread-local top-3 of x = logits * repeat_penality
        float v[3]  = {NEG, NEG, NEG};
        int   ii[3] = {0x7fffffff, 0x7fffffff, 0x7fffffff};
        for (int t = tid; t < VOCAB_; t += TPB_A)
            ins3(lb[t] * rb[t], t, v, ii);

        sv[tid*3+0] = v[0]; sv[tid*3+1] = v[1]; sv[tid*3+2] = v[2];
        si[tid*3+0] = ii[0]; si[tid*3+1] = ii[1]; si[tid*3+2] = ii[2];
        __syncthreads();

        // log-tree merge of 1024 sorted 3-lists
        for (int st = TPB_A / 2; st > 0; st >>= 1) {
            if (tid < st) {
                float av[3] = {sv[tid*3], sv[tid*3+1], sv[tid*3+2]};
                int   ai[3] = {si[tid*3], si[tid*3+1], si[tid*3+2]};
                const int o = (tid + st) * 3;
                float bv2[3] = {sv[o], sv[o+1], sv[o+2]};
                int   bi2[3] = {si[o], si[o+1], si[o+2]};
                merge3(av, ai, bv2, bi2);
                sv[tid*3] = av[0]; sv[tid*3+1] = av[1]; sv[tid*3+2] = av[2];
                si[tid*3] = ai[0]; si[tid*3+1] = ai[1]; si[tid*3+2] = ai[2];
            }
            __syncthreads();
        }

        // pass 2: stable softmax denominator (max == sv[0])
        const float M = sv[0];
        float acc = 0.f;
        for (int t = tid; t < VOCAB_; t += TPB_A)
            acc += __expf(lb[t] * rb[t] - M);
        s_red[tid] = acc;
        __syncthreads();
        for (int st = TPB_A / 2; st > 0; st >>= 1) {
            if (tid < st) s_red[tid] += s_red[tid + st];
            __syncthreads();
        }
        if (tid == 0) {
            float logZ = M + logf(s_red[0]);
            for (int k = 0; k < TOPK_; ++k) { btv[b][k] = sv[k] - logZ; bti[b][k] = si[k]; }
        }
        __syncthreads();
    }

    if (tid == 0) {
        float cur[9];
        for (int b = 0; b < 3; ++b)
            for (int k = 0; k < 3; ++k)
                cur[b*3+k] = btv[b][k] + prev[b];
        bool used[9] = {false,false,false,false,false,false,false,false,false};
        for (int j = 0; j < 3; ++j) {
            int best = 0; float bv = NEG; bool f = false;
            for (int m = 0; m < 9; ++m)            // ascending scan + strict '>'
                if (!used[m] && (!f || cur[m] > bv)) { f = true; bv = cur[m]; best = m; }
            used[best] = true;
            int bsel = best / 3, kk = best - 3 * bsel;
            s_bi[j] = bsel; s_tok[j] = bti[bsel][kk];
            ws[j] = bsel; ws[3 + j] = s_tok[j];
            out[O_PROB + j] = bv;
            out[O_TOK  + j] = (float)s_tok[j];
        }
        out[O_MAXI] = (float)s_tok[0];
    }
    __syncthreads();

    // save_id_out = concat(save_id[beam_index], tok) -> (3, 129)
    for (int idx = tid; idx < BEAM_ * (HIST_ + 1); idx += TPB_A) {
        int i = idx / (HIST_ + 1), c = idx - (HIST_ + 1) * i;
        int val = (c < HIST_) ? save_id[s_bi[i] * HIST_ + c] : s_tok[i];
        out[O_SAVE + idx] = (float)val;
    }
}

// ---------------------------------------------------------------------------
// Kernel B: repeat_penality[beam_index] gather via CDNA5 async Global<->LDS
// path (ASYNCcnt-tracked, no VGPR round-trip). tok fix-up happens in kernel C.
// ---------------------------------------------------------------------------
#define TPB_B 256
#define KSLOT 8
static constexpr long RP4 = (long)BEAM_ * VOCAB_ / 4;   // 24000 float4s

__global__ __launch_bounds__(TPB_B) void rp_async_gather_kernel(
    const float* __restrict__ rp_in, float* __restrict__ out,
    const int* __restrict__ ws)
{
    __shared__ __attribute__((aligned(16))) unsigned char lbuf[TPB_B * KSLOT * 16];
    const int tid = threadIdx.x;
    const unsigned lds0 = (unsigned)(uintptr_t)(&lbuf[0]);

    const int bi0 = ws[0], bi1 = ws[1], bi2 = ws[2];

    unsigned long long gdst[KSLOT];
    bool ok[KSLOT];

#pragma unroll
    for (int s = 0; s < KSLOT; ++s) {
        long e = (long)blockIdx.x * (TPB_B * KSLOT) + s * TPB_B + tid;
        ok[s] = (e < RP4);
        if (ok[s]) {
            int  row  = (int)(e / (VOCAB_ / 4));     // 8000 float4s per row
            long v4   = e - (long)row * (VOCAB_ / 4);
            int  bsel = (row == 0) ? bi0 : ((row == 1) ? bi1 : bi2);
            const float* sp = rp_in + (long)bsel * VOCAB_ + v4 * 4;
            float*       dp = out + O_RP + (long)row * VOCAB_ + v4 * 4;
            gdst[s] = (unsigned long long)(uintptr_t)dp;
            unsigned long long gsrc = (unsigned long long)(uintptr_t)sp;
            unsigned lo = lds0 + (unsigned)((tid * KSLOT + s) * 16);
            asm volatile("global_load_async_to_lds_b128 %0, %1, off"
                         :: "v"(lo), "v"(gsrc) : "memory");
        }
    }
    asm volatile("s_wait_asynccnt 0x0" ::: "memory");   // loads landed in LDS
#pragma unroll
    for (int s = 0; s < KSLOT; ++s) {
        if (ok[s]) {
            unsigned lo = lds0 + (unsigned)((tid * KSLOT + s) * 16);
            asm volatile("global_store_async_from_lds_b128 %0, %1, off"
                         :: "v"(gdst[s]), "v"(lo) : "memory");
        }
    }
    asm volatile("s_wait_asynccnt 0x0" ::: "memory");   // stores drained
}

// ---------------------------------------------------------------------------
// Kernel C: kv[:, beam_index] gather-copy at HBM rate. 806 MB of streaming
// traffic -> non-temporal hints both ways (doesn't fit 192 MB L2; avoid
// rinsing it). 8 x b128 in flight per lane, 12288 blocks. Also applies the
// rp tok fix-up (stream-ordered after kernel B).
// ---------------------------------------------------------------------------
#define TPB_C 256
#define UNR   8
static constexpr long CH4 = (long)H_ * S_ * D_ / 4;                 // 524288 float4 per (l,beam)
static constexpr int  BLK_PER_CHUNK = (int)(CH4 / (TPB_C * UNR));   // 256
static_assert(BLK_PER_CHUNK == 256, "shift below assumes 256 blocks/chunk");
static_assert(CH4 % (TPB_C * UNR) == 0, "exact tiling");

__global__ __launch_bounds__(TPB_C) void kv_gather_kernel(
    const v4f* __restrict__ src, float* __restrict__ outf,
    const float* __restrict__ rp_in, const float* __restrict__ pen,
    const int* __restrict__ ws)
{
    v4f* dst = (v4f*)outf;
    const int tid = threadIdx.x;

    int chunk = blockIdx.x >> 8;                 // / BLK_PER_CHUNK
    int bw    = blockIdx.x & (BLK_PER_CHUNK - 1);
    int l = chunk / 3, i = chunk - 3 * l;
    int bsel = ws[i];

    long srcBase = (long)(l * 3 + bsel) * CH4 + (long)bw * (TPB_C * UNR);
    long dstBase = (long)blockIdx.x * (TPB_C * UNR);

    v4f r[UNR];
#pragma unroll
    for (int s = 0; s < UNR; ++s)
        r[s] = __builtin_nontemporal_load(&src[srcBase + s * TPB_C + tid]);
#pragma unroll
    for (int s = 0; s < UNR; ++s)
        __builtin_nontemporal_store(r[s], &dst[dstBase + s * TPB_C + tid]);

    // rp.at[i, tok[i]] *= penality (recomputed from input -> idempotent/deterministic)
    if (blockIdx.x == 0 && tid < BEAM_) {
        int b = ws[tid], t = ws[3 + tid];
        outf[O_RP + (long)tid * VOCAB_ + t] = rp_in[(long)b * VOCAB_ + t] * pen[0];
    }
}

// ---------------------------------------------------------------------------
extern "C" void kernel_launch(void* const* d_in, const int* in_sizes, int n_in,
                              void* d_out, int out_size, void* d_ws, size_t ws_size,
                              hipStream_t stream) {
    const float* kv      = (const float*)d_in[0];
    const float* logits  = (const float*)d_in[1];
    const int*   save_id = (const int*)d_in[2];
    const float* rp_in   = (const float*)d_in[3];
    const float* prev    = (const float*)d_in[4];
    // d_in[5] = batch_indices (arange(3)) -- identity, unused
    const float* pen     = (const float*)d_in[6];
    (void)in_sizes; (void)n_in; (void)out_size; (void)ws_size;

    float* out = (float*)d_out;
    int*   ws  = (int*)d_ws;   // [0..2]=beam_index, [3..5]=tok

    beam_topk_kernel<<<1, TPB_A, 0, stream>>>(logits, save_id, rp_in, prev, out, ws);

    int blocksB = (int)((RP4 + (long)TPB_B * KSLOT - 1) / ((long)TPB_B * KSLOT)); // 12
    rp_async_gather_kernel<<<blocksB, TPB_B, 0, stream>>>(rp_in, out, ws);

    kv_gather_kernel<<<L_ * BEAM_ * BLK_PER_CHUNK, TPB_C, 0, stream>>>(
        (const v4f*)kv, out, rp_in, pen, ws);
}